// LSTM_9423158247923
// MI455X (gfx1250) — compile-verified
//
#include <hip/hip_runtime.h>
#include <math.h>

// Problem sizes (match reference)
#define B_SZ 256
#define T_SZ 512
#define F_SZ 512
#define H_SZ 512
#define G4H  2048   // 4*H

typedef float v2f __attribute__((ext_vector_type(2)));
typedef float v8f __attribute__((ext_vector_type(8)));

// ---------------------------------------------------------------------------
// Init: zero the two h-broadcast buffers and the two grid-sync counters.
// Must run every launch (workspace is poisoned / left dirty between calls).
// ---------------------------------------------------------------------------
__global__ void lstm_init(float* hb1, float* hb2, int* cnts) {
  int t = threadIdx.x;
  for (int i = t; i < H_SZ; i += 256) { hb1[i] = 0.f; hb2[i] = 0.f; }
  if (t < 2) cnts[t] = 0;
}

// ---------------------------------------------------------------------------
// Z = A(M x K, row stride lda) * W(K x N) + bias  via V_WMMA_F32_16X16X4_F32.
// One 16x16 tile per wave, 4 waves (64 output cols) per block.
// ISA lane layouts (CDNA5 ISA):
//   A 16x4 : lane l, vgpr v -> (M = l&15, K = v + 2*(l>>4))
//   B 4x16 : lane l, vgpr v -> (K = v + 2*(l>>4), N = l&15)
//   C/D    : lane l, vgpr v -> (M = v + 8*(l>>4), N = l&15)
// ---------------------------------------------------------------------------
__global__ void gemm_bias_wmma(const float* __restrict__ A, size_t lda,
                               const float* __restrict__ W,
                               const float* __restrict__ bias,
                               float* __restrict__ Z, int K, int N) {
  const int lane = threadIdx.x & 31;
  const int wv   = threadIdx.x >> 5;          // wave in block: 0..3
  const int l16  = lane & 15;
  const int hi   = lane >> 4;                 // 0 or 1
  const int n0   = (blockIdx.x * 4 + wv) * 16;
  const int m0   = blockIdx.y * 16;

  v8f acc;
  const float bv = bias[n0 + l16];            // bias depends only on column
  #pragma unroll
  for (int v = 0; v < 8; ++v) acc[v] = bv;

  const float* Arow = A + (size_t)(m0 + l16) * lda;
  for (int k0 = 0; k0 < K; k0 += 4) {
    v2f a, b;
    a[0] = Arow[k0 + 2 * hi + 0];
    a[1] = Arow[k0 + 2 * hi + 1];
    b[0] = W[(size_t)(k0 + 2 * hi + 0) * N + n0 + l16];
    b[1] = W[(size_t)(k0 + 2 * hi + 1) * N + n0 + l16];
    acc = __builtin_amdgcn_wmma_f32_16x16x4_f32(
        /*neg_a=*/false, a, /*neg_b=*/false, b,
        /*c_mod=*/(short)0, acc, /*reuse_a=*/false, /*reuse_b=*/false);
  }

  #pragma unroll
  for (int v = 0; v < 8; ++v)
    Z[(size_t)(m0 + v + 8 * hi) * N + n0 + l16] = acc[v];
}

// ---------------------------------------------------------------------------
// Persistent LSTM scan over b = 0..255 for the single live chain (t = T-1).
//   z[b] (4H) is precomputed x@Wi + b.  step: z += h@Wh; gates i,f,g,o.
// 128 WGs x 256 threads. WG g owns hidden units [4g, 4g+4): 16 gate columns
// of Wh cached in LDS (16*512*4B = 32 KB), h broadcast via global + counter
// barrier with acquire/release atomics (agent scope).
// ---------------------------------------------------------------------------
#define SCAN_G   128
#define SCAN_TPB 256
#define HPW 4            // hidden units per workgroup
#define COLS 16          // 4 gates * HPW

__global__ void lstm_scan(const float* __restrict__ Z,   // (B, 4H)
                          const float* __restrict__ Wh,  // (H, 4H) row-major
                          float* __restrict__ hbuf,      // (H), zeroed
                          float* __restrict__ hout,      // (B, H)
                          int* __restrict__ cnt) {
  __shared__ __align__(16) float Whs[COLS * H_SZ];  // 32 KB
  __shared__ __align__(16) float hs[H_SZ];          // 2 KB
  __shared__ float dots[COLS];

  const int tid  = threadIdx.x;
  const int wg   = blockIdx.x;
  const int lc   = tid >> 4;     // local column 0..15  (gate-major: q*4+k)
  const int part = tid & 15;     // 32-element chunk of the 512-dot
  const int q    = lc >> 2;
  const int kk   = lc & 3;
  const int gcol = q * H_SZ + wg * HPW + kk;   // column in (H,4H) Wh

  // One-time: cache our 16 Wh columns into LDS, column-major.
  for (int i = 0; i < 32; ++i) {
    int e = part * 32 + i;
    Whs[lc * H_SZ + e] = Wh[(size_t)e * G4H + gcol];
  }
  float creg = 0.f;              // cell state, valid for tid < HPW
  __syncthreads();

  for (int b = 0; b < B_SZ; ++b) {
    // broadcast h into LDS (2 floats/thread)
    hs[tid]       = hbuf[tid];
    hs[tid + 256] = hbuf[tid + 256];
    __syncthreads();

    // partial dot: 32 elements per thread, vectorized from LDS
    float acc = 0.f;
    const float4* wp = (const float4*)&Whs[lc * H_SZ + part * 32];
    const float4* hp = (const float4*)&hs[part * 32];
    #pragma unroll
    for (int i = 0; i < 8; ++i) {
      float4 w4 = wp[i], h4 = hp[i];
      acc += w4.x * h4.x + w4.y * h4.y + w4.z * h4.z + w4.w * h4.w;
    }
    // reduce over the 16 "part" lanes (stays within each 16-lane half-wave)
    acc += __shfl_xor(acc, 1);
    acc += __shfl_xor(acc, 2);
    acc += __shfl_xor(acc, 4);
    acc += __shfl_xor(acc, 8);
    if (part == 0) dots[lc] = acc;
    __syncthreads();

    if (tid < HPW) {
      const int gj = wg * HPW + tid;
      const float* zr = Z + (size_t)b * G4H;
      float zi = zr[0 * H_SZ + gj] + dots[0 * HPW + tid];
      float zf = zr[1 * H_SZ + gj] + dots[1 * HPW + tid];
      float zg = zr[2 * H_SZ + gj] + dots[2 * HPW + tid];
      float zo = zr[3 * H_SZ + gj] + dots[3 * HPW + tid];
      float ig = 1.f / (1.f + expf(-zi));
      float fg = 1.f / (1.f + expf(-zf));
      float og = 1.f / (1.f + expf(-zo));
      float gg = tanhf(zg);
      creg = fg * creg + ig * gg;
      float hv = og * tanhf(creg);
      hbuf[gj] = hv;                              // broadcast for next step
      hout[(size_t)b * H_SZ + gj] = hv;           // ys[b] output
    }
    __syncthreads();

    // grid barrier (monotonic counter; counters zeroed each launch)
    if (tid == 0) {
      __threadfence();
      __hip_atomic_fetch_add(cnt, 1, __ATOMIC_RELEASE, __HIP_MEMORY_SCOPE_AGENT);
      const int target = (b + 1) * SCAN_G;
      while (__hip_atomic_load(cnt, __ATOMIC_ACQUIRE, __HIP_MEMORY_SCOPE_AGENT) < target) {
        __builtin_amdgcn_s_sleep(2);
      }
      __threadfence();
    }
    __syncthreads();
  }
}

// ---------------------------------------------------------------------------
// out(256,10) = h2(256,512) @ Wd(512,10) + bd
// ---------------------------------------------------------------------------
__global__ void dense_out_k(const float* __restrict__ h2,
                            const float* __restrict__ Wd,
                            const float* __restrict__ bd,
                            float* __restrict__ out) {
  int idx = blockIdx.x * blockDim.x + threadIdx.x;
  if (idx >= B_SZ * 10) return;
  int r = idx / 10, o = idx % 10;
  float s = bd[o];
  const float* hr = h2 + (size_t)r * H_SZ;
  for (int k = 0; k < H_SZ; ++k) s += hr[k] * Wd[k * 10 + o];
  out[idx] = s;
}

// ---------------------------------------------------------------------------
extern "C" void kernel_launch(void* const* d_in, const int* in_sizes, int n_in,
                              void* d_out, int out_size, void* d_ws, size_t ws_size,
                              hipStream_t stream) {
  (void)in_sizes; (void)n_in; (void)out_size; (void)ws_size;
  const float* x   = (const float*)d_in[0];
  const float* Wi1 = (const float*)d_in[1];
  const float* Wh1 = (const float*)d_in[2];
  const float* b1  = (const float*)d_in[3];
  const float* Wi2 = (const float*)d_in[4];
  const float* Wh2 = (const float*)d_in[5];
  const float* b2  = (const float*)d_in[6];
  const float* Wd  = (const float*)d_in[7];
  const float* bd  = (const float*)d_in[8];
  float* out = (float*)d_out;

  // workspace layout (~5.3 MB)
  float* ws  = (float*)d_ws;
  float* z1  = ws;                          // 256*2048
  float* z2  = z1 + (size_t)B_SZ * G4H;     // 256*2048
  float* h1  = z2 + (size_t)B_SZ * G4H;     // 256*512
  float* h2  = h1 + (size_t)B_SZ * H_SZ;    // 256*512
  float* hb1 = h2 + (size_t)B_SZ * H_SZ;    // 512
  float* hb2 = hb1 + H_SZ;                  // 512
  int*   cnts = (int*)(hb2 + H_SZ);         // 2 counters

  lstm_init<<<1, 256, 0, stream>>>(hb1, hb2, cnts);

  dim3 gg(G4H / 64, B_SZ / 16);   // 32 x 16 blocks, 128 threads (4 waves)
  // layer 1 input GEMM on the single live slice x[:, T-1, :]
  gemm_bias_wmma<<<gg, 128, 0, stream>>>(x + (size_t)(T_SZ - 1) * F_SZ,
                                         (size_t)T_SZ * F_SZ,
                                         Wi1, b1, z1, F_SZ, G4H);
  lstm_scan<<<SCAN_G, SCAN_TPB, 0, stream>>>(z1, Wh1, hb1, h1, cnts + 0);

  gemm_bias_wmma<<<gg, 128, 0, stream>>>(h1, (size_t)H_SZ,
                                         Wi2, b2, z2, H_SZ, G4H);
  lstm_scan<<<SCAN_G, SCAN_TPB, 0, stream>>>(z2, Wh2, hb2, h2, cnts + 1);

  dense_out_k<<<(B_SZ * 10 + 255) / 256, 256, 0, stream>>>(h2, Wd, bd, out);
}